// HmmModel_60962765800094
// MI455X (gfx1250) — compile-verified
//
#include <hip/hip_runtime.h>

// HMM count-statistics (histogram scatter-add) for MI455X / gfx1250.
//
// inputs (setup_inputs order):
//   d_in[0] words       int32 [8192*512]
//   d_in[1] labels      int32 [8192*512]
//   d_in[2] trans_count f32   [65*65]
//   d_in[3] emit_count  f32   [500000*64]
// output: concat(trans_count', emit_count') = 4225 + 32,000,000 floats.
//
// Strategy (memory-bound; zero useful FLOPs):
//  * init d_out = inputs with NT loads (don't evict L2) + vectorized stores
//  * emit_count (128 MB) fits in the 192 MB L2 -> random global_atomic_add_f32
//    mostly resolve in L2; keep all stores/atomics RT, all streams NT.
//  * trans_count (4225 bins, extreme contention) -> per-block LDS histogram
//    (ds_add_u32, 16.9 KB of the 320 KB/WGP LDS), one global flush per block.

#define N_WORDS   500000
#define N_LABELS  64
#define TC_DIM    (N_LABELS + 1)                   // 65
#define TC_SIZE   (TC_DIM * TC_DIM)                // 4225
#define SEQ       512
#define NTOK      (8192u * 512u)                   // 4,194,304
#define EC_SIZE   (500000u * 64u)                  // 32,000,000
#define EC_NVEC   ((EC_SIZE - 4u) / 4u)            // 7,999,999 float4 in bulk

// ---------------------------------------------------------------------------
// Kernel 1a: copy the 4225-float transition block.
// ---------------------------------------------------------------------------
__global__ void hmm_copy_tc(const float* __restrict__ tc_in,
                            float* __restrict__ out) {
    unsigned i = blockIdx.x * blockDim.x + threadIdx.x;
    if (i < TC_SIZE) out[i] = __builtin_nontemporal_load(tc_in + i);
}

// ---------------------------------------------------------------------------
// Kernel 1b: copy the 32M-float emission block.
// Output region starts at float offset 4225 (byte 16900 == 4 mod 16):
// peel 3 head + 1 tail scalars; bulk is 16B-aligned float4 stores.
// Input reads are NT (b32) so the 128 MB stream does not evict the freshly
// written output lines from L2 (which the atomic phase wants resident).
// ---------------------------------------------------------------------------
__global__ void hmm_copy_ec(const float* __restrict__ ec_in,
                            float* __restrict__ out) {
    float* __restrict__ out_ec = out + TC_SIZE;
    const unsigned tid    = blockIdx.x * blockDim.x + threadIdx.x;
    const unsigned stride = gridDim.x * blockDim.x;

    if (tid == 0) {
        out_ec[0] = __builtin_nontemporal_load(ec_in + 0);
        out_ec[1] = __builtin_nontemporal_load(ec_in + 1);
        out_ec[2] = __builtin_nontemporal_load(ec_in + 2);
        out_ec[EC_SIZE - 1] = __builtin_nontemporal_load(ec_in + (EC_SIZE - 1));
    }

    for (unsigned k = tid; k < EC_NVEC; k += stride) {
        const unsigned idx = 3u + 4u * k;          // input side: 4B-aligned only
        float4 v;
        v.x = __builtin_nontemporal_load(ec_in + idx + 0);
        v.y = __builtin_nontemporal_load(ec_in + idx + 1);
        v.z = __builtin_nontemporal_load(ec_in + idx + 2);
        v.w = __builtin_nontemporal_load(ec_in + idx + 3);
        *reinterpret_cast<float4*>(out_ec + idx) = v;   // 16B-aligned b128 store
    }
}

// ---------------------------------------------------------------------------
// Kernel 2: scatter-add counts.
//  - previous label via wave32 shuffle: lanes cover consecutive tokens of one
//    row (SEQ % 32 == 0, wave base % 32 == 0), so only lane 0 reads memory.
//  - trans_count -> LDS histogram (ds_add_u32 no-return), flushed once.
//  - emit_count  -> no-return global_atomic_add_f32 (low contention, L2-hot).
//  - all 32-bit indices -> SGPR-base + VGPR-offset addressing, minimal VALU.
// ---------------------------------------------------------------------------
__global__ void hmm_count(const int* __restrict__ words,
                          const int* __restrict__ labels,
                          float* __restrict__ out) {
    __shared__ unsigned tc_lds[TC_SIZE];
    for (unsigned k = threadIdx.x; k < TC_SIZE; k += blockDim.x) tc_lds[k] = 0u;
    __syncthreads();

    float* __restrict__ tc_out = out;
    float* __restrict__ ec_out = out + TC_SIZE;

    const unsigned lane   = threadIdx.x & 31u;
    unsigned       i      = blockIdx.x * blockDim.x + threadIdx.x;
    const unsigned stride = gridDim.x * blockDim.x;   // multiple of 32

    for (; i < NTOK; i += stride) {
        int w = __builtin_nontemporal_load(words + i);   // streamed once: NT
        int t = __builtin_nontemporal_load(labels + i);

        // Previous label from neighbor lane; lane 0 loads from memory
        // (or uses sentinel N_LABELS at sequence start j==0).
        int pre = __shfl_up(t, 1, 32);
        if (lane == 0u) {
            const unsigned j = i & (SEQ - 1u);
            pre = (j == 0u) ? N_LABELS : labels[i - 1u];
        }

        if (w != 0) {                                     // valid token
            unsigned uw = (unsigned)w;
            if (uw >= (unsigned)N_WORDS) uw = 1u;         // UNK clamp
            // trans_count[t, pre] += 1  -> ds_add_u32 (no return)
            atomicAdd(&tc_lds[(unsigned)t * TC_DIM + (unsigned)pre], 1u);
            // emit_count[w, t] += 1     -> global_atomic_add_f32 (no return)
            atomicAdd(&ec_out[uw * (unsigned)N_LABELS + (unsigned)t], 1.0f);
        }
    }

    __syncthreads();
    // Flush block-private transition histogram to global.
    for (unsigned k = threadIdx.x; k < TC_SIZE; k += blockDim.x) {
        const unsigned c = tc_lds[k];
        if (c) atomicAdd(&tc_out[k], (float)c);
    }
}

// ---------------------------------------------------------------------------
extern "C" void kernel_launch(void* const* d_in, const int* in_sizes, int n_in,
                              void* d_out, int out_size, void* d_ws, size_t ws_size,
                              hipStream_t stream) {
    const int*   words  = (const int*)d_in[0];
    const int*   labels = (const int*)d_in[1];
    const float* tc_in  = (const float*)d_in[2];
    const float* ec_in  = (const float*)d_in[3];
    float*       out    = (float*)d_out;

    // 1a: 4225 floats.
    hmm_copy_tc<<<(TC_SIZE + 255) / 256, 256, 0, stream>>>(tc_in, out);

    // 1b: ~8M float4 copies; 1M threads, ~8 vec-iterations each.
    hmm_copy_ec<<<4096, 256, 0, stream>>>(ec_in, out);

    // 2: 1M threads, exactly 4 tokens each (4,194,304 / 1,048,576 == 4).
    hmm_count<<<4096, 256, 0, stream>>>(words, labels, out);
}